// Hopfield_28200755266000
// MI455X (gfx1250) — compile-verified
//
#include <hip/hip_runtime.h>
#include <hip/hip_bf16.h>

#define VOC 100000
#define BAT 256
#define HID 128

typedef float v2f __attribute__((ext_vector_type(2)));
typedef float v8f __attribute__((ext_vector_type(8)));

static __device__ __forceinline__ v8f wmma4(v2f a, v2f b, v8f c) {
  // D = A(16x4 f32) * B(4x16 f32) + C(16x16 f32)
  return __builtin_amdgcn_wmma_f32_16x16x4_f32(false, a, false, b, (short)0, c,
                                               false, false);
}

static __device__ __forceinline__ v8f v8zero() {
  v8f z = {0.f, 0.f, 0.f, 0.f, 0.f, 0.f, 0.f, 0.f};
  return z;
}

// ---------------------------------------------------------------- zero init
__global__ void k_zero(float* __restrict__ p, int n) {
  int i = blockIdx.x * blockDim.x + threadIdx.x;
  if (i < n) p[i] = 0.f;
}

// --------------------------------------------------- part1 = context @ cw.T
// split-K: 125 WGs x 800 K each. WG = 512 threads = 16 waves.
// wave w owns M-tile w (rows 16w..16w+15) x all 8 N-tiles.
__global__ __launch_bounds__(512) void k_part1(const float* __restrict__ ctx,
                                               const float* __restrict__ cw,
                                               float* __restrict__ part1) {
  __shared__ float sA[256][36];  // context chunk [m][k], stride 36 (16B-aligned rows)
  __shared__ float sB[128][36];  // cw chunk [n][k]
  const int t = threadIdx.x;
  const int wave = t >> 5;
  const int lane = t & 31;
  const int l16 = lane & 15;
  const int lh = lane >> 4;  // K half select
  const int k0wg = blockIdx.x * 800;
  const int q = t & 7;
  const int r0 = t >> 3;  // 0..63

  v8f acc[8];
#pragma unroll
  for (int nt = 0; nt < 8; ++nt) acc[nt] = v8zero();

  for (int ch = 0; ch < 25; ++ch) {
    const int kc = k0wg + ch * 32;
    __syncthreads();
#pragma unroll
    for (int i = 0; i < 4; ++i) {  // A: 256 rows x 32 K
      const int m = r0 + 64 * i;
      const float4 v = *(const float4*)(ctx + (size_t)m * VOC + kc + q * 4);
      *(float4*)&sA[m][q * 4] = v;
    }
#pragma unroll
    for (int i = 0; i < 2; ++i) {  // B: 128 rows x 32 K
      const int n = r0 + 64 * i;
      const float4 v = *(const float4*)(cw + (size_t)n * VOC + kc + q * 4);
      *(float4*)&sB[n][q * 4] = v;
    }
    __syncthreads();
#pragma unroll
    for (int s = 0; s < 8; ++s) {  // 8 micro-steps of K=4
      const int kk = s * 4 + lh * 2;
      // issue ALL fragment loads first so they form one LDS clause with a
      // single s_wait_dscnt, then run the 8 WMMAs back-to-back
      const v2f a = *(const v2f*)&sA[wave * 16 + l16][kk];
      v2f b[8];
#pragma unroll
      for (int nt = 0; nt < 8; ++nt) b[nt] = *(const v2f*)&sB[nt * 16 + l16][kk];
#pragma unroll
      for (int nt = 0; nt < 8; ++nt) acc[nt] = wmma4(a, b[nt], acc[nt]);
    }
  }
  // accumulate split-K partials
#pragma unroll
  for (int nt = 0; nt < 8; ++nt) {
#pragma unroll
    for (int r = 0; r < 8; ++r) {
      const int m = wave * 16 + r + lh * 8;
      const int n = nt * 16 + l16;
      atomicAdd(&part1[m * HID + n], acc[nt][r]);
    }
  }
}

// --------------------------------------------------------- G = tw @ tw.T
__global__ __launch_bounds__(256) void k_gram(const float* __restrict__ tw,
                                              float* __restrict__ G) {
  __shared__ float sT[128][36];
  const int t = threadIdx.x;
  const int wave = t >> 5;  // 0..7 = M-tile
  const int lane = t & 31;
  const int l16 = lane & 15;
  const int lh = lane >> 4;
  const int k0wg = blockIdx.x * 800;
  const int q = t & 7;
  const int r0 = t >> 3;  // 0..31

  v8f acc[8];
#pragma unroll
  for (int nt = 0; nt < 8; ++nt) acc[nt] = v8zero();

  for (int ch = 0; ch < 25; ++ch) {
    const int kc = k0wg + ch * 32;
    __syncthreads();
#pragma unroll
    for (int i = 0; i < 4; ++i) {
      const int n = r0 + 32 * i;
      const float4 v = *(const float4*)(tw + (size_t)n * VOC + kc + q * 4);
      *(float4*)&sT[n][q * 4] = v;
    }
    __syncthreads();
#pragma unroll
    for (int s = 0; s < 8; ++s) {
      const int kk = s * 4 + lh * 2;
      const v2f a = *(const v2f*)&sT[wave * 16 + l16][kk];
      v2f b[8];
#pragma unroll
      for (int nt = 0; nt < 8; ++nt) b[nt] = *(const v2f*)&sT[nt * 16 + l16][kk];
#pragma unroll
      for (int nt = 0; nt < 8; ++nt) acc[nt] = wmma4(a, b[nt], acc[nt]);
    }
  }
#pragma unroll
  for (int nt = 0; nt < 8; ++nt) {
#pragma unroll
    for (int r = 0; r < 8; ++r) {
      const int m = wave * 16 + r + lh * 8;
      const int n = nt * 16 + l16;
      atomicAdd(&G[m * HID + n], acc[nt][r]);
    }
  }
}

// ------------------------------------------- softmax recurrence (one step)
// block = one batch row; 128 threads = one hidden unit each; P updated in place
__global__ __launch_bounds__(128) void k_iter(const float* __restrict__ part1,
                                              const float* __restrict__ G,
                                              float* __restrict__ P,
                                              const float* __restrict__ b1,
                                              const float* __restrict__ b2,
                                              int it) {
  __shared__ float sp[128];
  __shared__ float sred[128];
  const int b = blockIdx.x, h = threadIdx.x;
  float logit = part1[b * HID + h] * b1[0];
  if (it > 0) {
    sp[h] = P[b * HID + h];
    __syncthreads();
    float dot = 0.f;
#pragma unroll 8
    for (int k = 0; k < HID; ++k) dot = fmaf(sp[k], G[k * HID + h], dot);
    logit = fmaf(b2[0], dot, logit);
    __syncthreads();
  }
  sred[h] = logit;
  __syncthreads();
  for (int s = 64; s > 0; s >>= 1) {
    if (h < s) sred[h] = fmaxf(sred[h], sred[h + s]);
    __syncthreads();
  }
  const float mx = sred[0];
  __syncthreads();
  const float e = __expf(logit - mx);
  sred[h] = e;
  __syncthreads();
  for (int s = 64; s > 0; s >>= 1) {
    if (h < s) sred[h] += sred[h + s];
    __syncthreads();
  }
  P[b * HID + h] = e / sred[0];
}

// ------------------------------- out = BatchNorm(P @ tw), 32 columns per WG
__global__ __launch_bounds__(256) void k_final(const float* __restrict__ P,
                                               const float* __restrict__ tw,
                                               const float* __restrict__ gam,
                                               const float* __restrict__ bet,
                                               float* __restrict__ out) {
  __shared__ float sP[256][36];    // p3 staged [m][k]; later reused as X[m][col]
  __shared__ float sBT[32][134];   // tw tile transposed [n][k]
  __shared__ float rsum[8][32];
  __shared__ float rsq[8][32];
  __shared__ float sScale[32];
  __shared__ float sShift[32];
  const int t = threadIdx.x;
  const int wave = t >> 5;
  const int lane = t & 31;
  const int l16 = lane & 15;
  const int lh = lane >> 4;
  const size_t jbase = (size_t)blockIdx.x * 32;

  {  // stage P: 256x128 floats, coalesced float4
    const int q = t & 31, m0 = t >> 5;
#pragma unroll
    for (int i = 0; i < 32; ++i) {
      const int m = m0 + 8 * i;
      const float4 v = *(const float4*)(P + m * HID + q * 4);
      *(float4*)&sP[m][q * 4] = v;
    }
    // stage tw tile transposed: coalesced over columns
    const int n = t & 31, kk0 = t >> 5;
#pragma unroll
    for (int i = 0; i < 16; ++i) {
      const int k = kk0 + 8 * i;
      sBT[n][k] = tw[(size_t)k * VOC + jbase + n];
    }
  }
  __syncthreads();

  // wave w: M-tiles {2w, 2w+1} x N-tiles {0,1}; K = 128 in 32 micro-steps
  v8f a00 = v8zero(), a01 = v8zero(), a10 = v8zero(), a11 = v8zero();
#pragma unroll
  for (int s = 0; s < 32; ++s) {
    const int kk = s * 4 + lh * 2;
    const v2f fa0 = *(const v2f*)&sP[(2 * wave) * 16 + l16][kk];
    const v2f fa1 = *(const v2f*)&sP[(2 * wave + 1) * 16 + l16][kk];
    const v2f fb0 = *(const v2f*)&sBT[l16][kk];
    const v2f fb1 = *(const v2f*)&sBT[16 + l16][kk];
    a00 = wmma4(fa0, fb0, a00);
    a01 = wmma4(fa0, fb1, a01);
    a10 = wmma4(fa1, fb0, a10);
    a11 = wmma4(fa1, fb1, a11);
  }
  __syncthreads();  // everyone done reading sP -> reuse as X

#pragma unroll
  for (int r = 0; r < 8; ++r) {
    const int mA = (2 * wave) * 16 + r + lh * 8;
    const int mB = (2 * wave + 1) * 16 + r + lh * 8;
    sP[mA][l16] = a00[r];
    sP[mA][16 + l16] = a01[r];
    sP[mB][l16] = a10[r];
    sP[mB][16 + l16] = a11[r];
  }
  __syncthreads();

  {  // batch statistics: 8 segments of 32 rows per column
    const int c = t & 31, seg = t >> 5;
    float sx = 0.f, sx2 = 0.f;
#pragma unroll 8
    for (int r = seg * 32; r < seg * 32 + 32; ++r) {
      const float x = sP[r][c];
      sx += x;
      sx2 = fmaf(x, x, sx2);
    }
    rsum[seg][c] = sx;
    rsq[seg][c] = sx2;
  }
  __syncthreads();
  if (t < 32) {
    float sx = 0.f, sx2 = 0.f;
#pragma unroll
    for (int s = 0; s < 8; ++s) {
      sx += rsum[s][t];
      sx2 += rsq[s][t];
    }
    const float mean = sx * (1.0f / 256.0f);
    const float var = sx2 * (1.0f / 256.0f) - mean * mean;  // biased
    const float inv = rsqrtf(var + 1e-5f);
    const float g = gam[jbase + t];
    sScale[t] = inv * g;
    sShift[t] = bet[jbase + t] - mean * inv * g;
  }
  __syncthreads();
  {  // coalesced store: 32 consecutive columns per wavefront row-pass
    const int c = t & 31, m0 = t >> 5;
#pragma unroll
    for (int i = 0; i < 32; ++i) {
      const int m = m0 + 8 * i;
      out[(size_t)m * VOC + jbase + c] = fmaf(sP[m][c], sScale[c], sShift[c]);
    }
  }
}

extern "C" void kernel_launch(void* const* d_in, const int* in_sizes, int n_in,
                              void* d_out, int out_size, void* d_ws,
                              size_t ws_size, hipStream_t stream) {
  const float* ctx = (const float*)d_in[0];   // [256,100000]
  const float* cw = (const float*)d_in[1];    // [128,100000]
  const float* tw = (const float*)d_in[2];    // [128,100000]
  const float* gam = (const float*)d_in[3];   // [100000]
  const float* bet = (const float*)d_in[4];   // [100000]
  const float* b1 = (const float*)d_in[5];    // [1]
  const float* b2 = (const float*)d_in[6];    // [1]
  float* out = (float*)d_out;                 // [256,100000]
  float* ws = (float*)d_ws;
  float* part1 = ws;          // 32768 floats
  float* G = ws + 32768;      // 16384 floats
  float* P = ws + 49152;      // 32768 floats

  k_zero<<<192, 256, 0, stream>>>(ws, 49152);           // part1 + G = 0
  k_part1<<<125, 512, 0, stream>>>(ctx, cw, part1);     // split-K WMMA
  k_gram<<<125, 256, 0, stream>>>(tw, G);               // split-K WMMA
  for (int it = 0; it < 3; ++it)
    k_iter<<<256, 128, 0, stream>>>(part1, G, P, b1, b2, it);
  k_final<<<3125, 256, 0, stream>>>(P, tw, gam, bet, out);  // GEMM + fused BN
}